// Castro2025Model_18597208391790
// MI455X (gfx1250) — compile-verified
//
#include <hip/hip_runtime.h>
#include <hip/hip_bf16.h>
#include <stdint.h>

// ---- problem constants (match reference) ----
#define T_TRIALS 1000
#define NB       8192       // batch
#define NP       1024       // param sets
#define NA       4          // arms
#define CHUNK    8          // trials staged per LDS buffer (1000/8 = 125 exact)
#define BB       64         // batch elements per block (256 threads, lane=arm)
#define NCHUNK   (T_TRIALS / CHUNK)
#define NBUF     4          // async pipeline depth (prefetch distance 3)
#define ER_DECAY 0.999f

// ---- CDNA5 async global->LDS staging (gfx1250) ----
__device__ __forceinline__ void async_ld_b32(unsigned lds_addr, const void* gaddr) {
    asm volatile("global_load_async_to_lds_b32 %0, %1, off"
                 :: "v"(lds_addr), "v"(gaddr) : "memory");
}
#define WAIT_ASYNC(n) asm volatile("s_wait_asynccnt %0" :: "i"(n) : "memory")

// ---- quad (4-lane) reductions via DPP quad_perm (pure VALU, no LDS) ----
__device__ __forceinline__ float qswap1(float v) {     // lane ^ 1
    int i = __float_as_int(v);
    return __int_as_float(__builtin_amdgcn_update_dpp(i, i, 0xB1, 0xF, 0xF, false));
}
__device__ __forceinline__ float qswap2(float v) {     // lane ^ 2
    int i = __float_as_int(v);
    return __int_as_float(__builtin_amdgcn_update_dpp(i, i, 0x4E, 0xF, 0xF, false));
}
__device__ __forceinline__ float qsum4(float v) {
    v += qswap1(v);
    v += qswap2(v);
    return v;
}
__device__ __forceinline__ float qmax4(float v) {
    v = fmaxf(v, qswap1(v));
    v = fmaxf(v, qswap2(v));
    return v;
}

__device__ __forceinline__ float clipf(float x, float lo, float hi) {
    return fminf(fmaxf(x, lo), hi);
}
__device__ __forceinline__ float softplusf(float x) {   // x already clipped to [-5,5]
    return log1pf(__expf(x));
}
__device__ __forceinline__ float sigmoidf(float x) {
    return __builtin_amdgcn_rcpf(1.0f + __expf(-x));
}

__global__ __launch_bounds__(256) void castro2025_scan_kernel(
    const float* __restrict__ params,    // (13, NP)
    const float* __restrict__ rewards,   // (T, NB)
    const int*   __restrict__ choices,   // (T, NB)
    const int*   __restrict__ pids,      // (NB,)
    float*       __restrict__ out)       // (T, NB, NA)
{
    // interleaved staging: entry 2*(tl*BB+bl) = choice, +1 = reward(bits)
    __shared__ int s_cr[NBUF][CHUNK * BB * 2];

    const int tid = threadIdx.x;          // 0..255
    const int arm = tid & 3;              // lane-in-quad = arm
    const int bl  = tid >> 2;             // chain within block 0..63
    const int b0  = blockIdx.x * BB;
    const int b   = b0 + bl;

    // ---- async loader mapping: thread i stages 2 columns of one chunk row ----
    const int lr = tid >> 5;              // chunk row 0..7
    const int lc = (tid & 31) * 2;        // column pair 0..62
    unsigned lbase[NBUF];
    #pragma unroll
    for (int u = 0; u < NBUF; ++u)
        lbase[u] = (unsigned)(uintptr_t)&s_cr[u][(lr * BB + lc) * 2];

    // stage chunk k into buffer k & (NBUF-1): 4 async b32 per thread
    auto stage = [&](int k) {
        const size_t g = (size_t)(k * CHUNK + lr) * NB + b0 + lc;
        const unsigned l = lbase[k & (NBUF - 1)];
        async_ld_b32(l,      choices + g);
        async_ld_b32(l + 4,  rewards + g);
        async_ld_b32(l + 8,  choices + g + 1);
        async_ld_b32(l + 12, rewards + g + 1);
    };

    // prologue: fill the pipeline (chunks 0..2) before parameter setup
    stage(0);
    stage(1);
    stage(2);

    // ---- per-chain parameter transforms (once; redundant across quad) ----
    const int pid = pids[b];
    float p[13];
    #pragma unroll
    for (int r = 0; r < 13; ++r)
        p[r] = clipf(params[r * NP + pid], -5.0f, 5.0f);

    const float beta_r = clipf(softplusf(p[0]), 0.01f, 20.0f);
    const float lapse  = clipf(sigmoidf(p[1]), 0.01f, 0.99f);
    const float prior  = clipf(softplusf(p[2]), 0.01f, 0.99f);
    float       er     = clipf(sigmoidf(p[3]), 0.01f, 0.99f);
    const float decay  = clipf(sigmoidf(p[4]), 0.01f, 0.99f);
    const float ab1    = p[5];
    const float ab2    = p[6];
    const float perv   = softplusf(p[7]);
    const float sw     = p[8];
    const float gam    = softplusf(p[10]);
    const float temp   = clipf(softplusf(p[11]) + 1e-6f, 1e-6f, 100.0f);
    const float beta_p = softplusf(p[12]);

    const float brt    = beta_r / temp;       // beta_r / temp folded
    const float oml    = 1.0f - lapse;
    const float lapse4 = 0.25f * lapse;       // lapse / A

    // ---- recurrent state (per lane: this arm's slice) ----
    float q    = prior;                        // q[arm]
    float cum  = 0.0f;                         // cum[arm]
    float lcum = 0.0f;                         // log1p(cum[arm]) cached
    float tsls = 0.0f;
    int   old_choice = -1;

    float*       outp    = out + (size_t)b * NA + arm;
    const size_t ostride = (size_t)NB * NA;

    for (int k = 0; k < NCHUNK; ++k) {
        const int buf = k & (NBUF - 1);

        // chunks in flight beyond k: up to 2 (k+1, k+2) at 4 asyncs each
        if      (k + 2 < NCHUNK) WAIT_ASYNC(8);
        else if (k + 1 < NCHUNK) WAIT_ASYNC(4);
        else                     WAIT_ASYNC(0);
        __syncthreads();          // chunk-k data globally visible; buffer (k+3)&3 free

        if (k + 3 < NCHUNK) stage(k + 3);

        #pragma unroll
        for (int tl = 0; tl < CHUNK; ++tl) {
            const int2  cr     = *(const int2*)&s_cr[buf][(tl * BB + bl) * 2];
            const int   choice = cr.x;
            const float reward = __int_as_float(cr.y);
            const bool  is_c   = (arm == choice);
            const bool  same   = (choice == old_choice);

            // q[choice] = reward - gam*(1-reward)
            const float target = reward - gam * (1.0f - reward);
            if (is_c) q = target;

            tsls = same ? (tsls + 1.0f) : 0.0f;
            er  *= ER_DECAY;
            if (is_c) { cum += 1.0f; lcum = __logf(1.0f + cum); }

            // exploration smoothing toward mean, then decay
            const float qm = 0.25f * qsum4(q);
            q = q + er * (qm - q);
            q *= decay;

            // softmax over the quad
            const float x  = brt * q + beta_p * lcum;
            const float mx = qmax4(x);
            const float e  = __expf(x - mx);
            const float S  = qsum4(e);
            const float prob = oml * (e * __builtin_amdgcn_rcpf(S)) + lapse4;

            float cl = __logf(prob);
            if (is_c)                       cl += same ? perv : sw;
            if (arm == old_choice)          cl += ab1;   // old_choice==-1 -> no lane
            if (arm == ((choice + 2) & 3))  cl += ab2;
            if (is_c)                       cl += __logf(tsls + 1.0f);

            __builtin_nontemporal_store(cl, outp);   // write-once 131MB stream
            outp += ostride;
            old_choice = choice;
        }
    }
}

extern "C" void kernel_launch(void* const* d_in, const int* in_sizes, int n_in,
                              void* d_out, int out_size, void* d_ws, size_t ws_size,
                              hipStream_t stream) {
    (void)in_sizes; (void)n_in; (void)out_size; (void)d_ws; (void)ws_size;
    const float* params  = (const float*)d_in[0];   // (13, 1024)
    const float* rewards = (const float*)d_in[1];   // (1000, 8192)
    const int*   choices = (const int*)  d_in[2];   // (1000, 8192)
    const int*   pids    = (const int*)  d_in[3];   // (8192,)
    float*       out     = (float*)d_out;           // (1000, 8192, 4)

    castro2025_scan_kernel<<<NB / BB, 256, 0, stream>>>(params, rewards, choices, pids, out);
}